// GDN_19971597926631
// MI455X (gfx1250) — compile-verified
//
#include <hip/hip_runtime.h>
#include <math.h>

typedef __attribute__((ext_vector_type(2))) float v2f;
typedef __attribute__((ext_vector_type(8))) float v8f;

#define CH 128
#define LDSP 132              // padded LDS row stride (floats): breaks 16-way bank conflicts
#define WAVES 4
#define MAX_BLOCKS 4096       // persistent grid; each block stages g exactly once

// reparameterization constants (exact from reference)
#define PED        1.4551915228366852e-11f   // 2^-36
#define GAMMA_BND  3.814697265625e-06f       // 2^-18
#define BETA_BND   1.0000072759536418e-03f   // sqrt(1e-6 + 2^-36)

__global__ __launch_bounds__(128) void gdn_wmma_f32(
    const float* __restrict__ x,
    const float* __restrict__ beta,
    const float* __restrict__ gamma,
    float* __restrict__ out,
    int nrows)
{
    __shared__ float gs[CH * LDSP];              // reparam'd gamma, row i = g[i][:]
    __shared__ float xs[WAVES * 16 * LDSP];      // per-wave raw x tiles (wave-private)

    const int tid   = threadIdx.x;
    const int lane  = tid & 31;
    const int wave  = tid >> 5;
    const int laneN = lane & 15;
    const int half  = lane >> 4;

    // ---- stage reparameterized g into LDS once per block (coalesced float4) ----
    {
        const float4* g4 = (const float4*)gamma;
        #pragma unroll
        for (int t = 0; t < 32; ++t) {
            int idx = t * 128 + tid;      // float4 index within 128x128
            int row = idx >> 5;           // 32 float4 per row
            int c4  = idx & 31;
            float4 v = g4[idx];
            v.x = fmaxf(v.x, GAMMA_BND); v.x = v.x * v.x - PED;
            v.y = fmaxf(v.y, GAMMA_BND); v.y = v.y * v.y - PED;
            v.z = fmaxf(v.z, GAMMA_BND); v.z = v.z * v.z - PED;
            v.w = fmaxf(v.w, GAMMA_BND); v.w = v.w * v.w - PED;
            *(float4*)&gs[row * LDSP + c4 * 4] = v;   // 528B row pitch: 16B aligned
        }
    }
    __syncthreads();   // only barrier in the kernel: gs is block-shared

    float* xw = &xs[wave * 16 * LDSP];
    const int ntiles = (nrows + 15) >> 4;
    const int tstep  = gridDim.x * WAVES;

    // ---- persistent grid-stride loop over 16-row tiles ----
    for (int tile = blockIdx.x * WAVES + wave; tile < ntiles; tile += tstep) {
        const int rowBase = tile * 16;

        // stage this wave's 16x128 x tile into its private LDS region (coalesced)
        #pragma unroll
        for (int t = 0; t < 16; ++t) {
            int r = rowBase + t;
            if (r > nrows - 1) r = nrows - 1;        // clamp (keeps EXEC full)
            const float4* xr = (const float4*)(x + (size_t)r * CH);
            float4 v = xr[lane];                      // 32 lanes x 16B = 512B/instr
            *(float4*)&xw[t * LDSP + lane * 4] = v;
        }

        // A fragments (x^2), loaded once, reused across all 8 N tiles.
        // A 16x4 fp32 layout: lane(M)=laneN; half=0 -> K={0,1}, half=1 -> K={2,3}
        v2f a[32];
        #pragma unroll
        for (int k = 0; k < 32; ++k) {
            v2f v = *(const v2f*)&xw[laneN * LDSP + k * 4 + 2 * half];
            v.x *= v.x;
            v.y *= v.y;
            a[k] = v;
        }

        // 8 N-tiles processed four at a time: four independent WMMA accumulator
        // chains push each chain's D->C dependency 4 issues back (fully pipelined).
        const size_t outBase = (size_t)rowBase * CH;
        for (int nt = 0; nt < 8; nt += 4) {
            const int nCol0 = nt * 16 + laneN;
            const int nCol1 = nCol0 + 16;
            const int nCol2 = nCol0 + 32;
            const int nCol3 = nCol0 + 48;

            float bv0 = fmaxf(beta[nCol0], BETA_BND); bv0 = bv0 * bv0 - PED;
            float bv1 = fmaxf(beta[nCol1], BETA_BND); bv1 = bv1 * bv1 - PED;
            float bv2 = fmaxf(beta[nCol2], BETA_BND); bv2 = bv2 * bv2 - PED;
            float bv3 = fmaxf(beta[nCol3], BETA_BND); bv3 = bv3 * bv3 - PED;

            v8f acc0 = {};
            v8f acc1 = {};
            v8f acc2 = {};
            v8f acc3 = {};
            #pragma unroll
            for (int k = 0; k < 32; ++k) {
                // B[k'][n] = g[n][k'] -> row nCol of gs; half selects K pair
                const int co = k * 4 + 2 * half;
                v2f b0 = *(const v2f*)&gs[nCol0 * LDSP + co];
                v2f b1 = *(const v2f*)&gs[nCol1 * LDSP + co];
                v2f b2 = *(const v2f*)&gs[nCol2 * LDSP + co];
                v2f b3 = *(const v2f*)&gs[nCol3 * LDSP + co];
                acc0 = __builtin_amdgcn_wmma_f32_16x16x4_f32(
                    false, a[k], false, b0, (short)0, acc0, false, false);
                acc1 = __builtin_amdgcn_wmma_f32_16x16x4_f32(
                    false, a[k], false, b1, (short)0, acc1, false, false);
                acc2 = __builtin_amdgcn_wmma_f32_16x16x4_f32(
                    false, a[k], false, b2, (short)0, acc2, false, false);
                acc3 = __builtin_amdgcn_wmma_f32_16x16x4_f32(
                    false, a[k], false, b3, (short)0, acc3, false, false);
            }

            // C/D layout: VGPR v -> M = v + 8*half, N = nCol
            #pragma unroll
            for (int v = 0; v < 8; ++v) {
                int r = v + 8 * half;
                bool ok = (rowBase + r < nrows);
                float s0 = acc0[v] + bv0;
                float s1 = acc1[v] + bv1;
                float s2 = acc2[v] + bv2;
                float s3 = acc3[v] + bv3;
                float x0 = xw[r * LDSP + nCol0];
                float x1 = xw[r * LDSP + nCol1];
                float x2 = xw[r * LDSP + nCol2];
                float x3 = xw[r * LDSP + nCol3];
                if (ok) {
                    size_t rb = outBase + (size_t)r * CH;
                    out[rb + nCol0] = x0 / sqrtf(s0);
                    out[rb + nCol1] = x1 / sqrtf(s1);
                    out[rb + nCol2] = x2 / sqrtf(s2);
                    out[rb + nCol3] = x3 / sqrtf(s3);
                }
            }
        }
    }
}

extern "C" void kernel_launch(void* const* d_in, const int* in_sizes, int n_in,
                              void* d_out, int out_size, void* d_ws, size_t ws_size,
                              hipStream_t stream) {
    const float* x     = (const float*)d_in[0];
    const float* beta  = (const float*)d_in[1];
    const float* gamma = (const float*)d_in[2];
    float* out = (float*)d_out;

    int nrows  = in_sizes[0] / CH;                 // 1,000,000
    int ntiles = (nrows + 15) / 16;                // 62,500 wave-tiles
    int blocks = (ntiles + WAVES - 1) / WAVES;     // one tile per wave if small
    if (blocks > MAX_BLOCKS) blocks = MAX_BLOCKS;  // persistent grid

    gdn_wmma_f32<<<blocks, 128, 0, stream>>>(x, beta, gamma, out, nrows);
}